// GNN_RL_37830071943833
// MI455X (gfx1250) — compile-verified
//
#include <hip/hip_runtime.h>
#include <hip/hip_bf16.h>

typedef __attribute__((ext_vector_type(2))) float v2f;
typedef __attribute__((ext_vector_type(8))) float v8f;

#define WG 256

// ---------------------------------------------------------------- utilities
__global__ void k_fill(float* __restrict__ p, int n, float v) {
  int i = blockIdx.x * blockDim.x + threadIdx.x;
  if (i < n) p[i] = v;
}

// deg[dst[e]] += edge_attr[e,0]   (deg pre-initialized to 1.0 = self loop)
__global__ void k_degree(const int* __restrict__ dst, const float* __restrict__ ea,
                         float* __restrict__ deg, int E) {
  int e = blockIdx.x * blockDim.x + threadIdx.x;
  if (e < E) atomicAdd(&deg[dst[e]], ea[(size_t)e * 4]);
}

// dis[i] = deg>0 ? rsqrt(deg) : 0   (in place)
__global__ void k_rsqrt(float* __restrict__ deg, int n) {
  int i = blockIdx.x * blockDim.x + threadIdx.x;
  if (i < n) {
    float d = deg[i];
    deg[i] = d > 0.0f ? rsqrtf(d) : 0.0f;
  }
}

// ---------------------------------------------------------- h1 = x @ W1 (5->16)
// One wave per 16 rows. Exact f32 WMMA for K=0..3, scalar FMA for K=4.
__global__ void k_xw1_wmma(const float* __restrict__ x, const float* __restrict__ W1,
                           float* __restrict__ h1, int N) {
  int lane = threadIdx.x & 31;
  int wave = (blockIdx.x * blockDim.x + threadIdx.x) >> 5;
  int base = wave * 16;
  if (base >= N) return;                       // uniform per-wave exit
  int half = lane >> 4;                        // 0: K=0,1  1: K=2,3
  int l16  = lane & 15;
  if (base + 16 <= N) {                        // uniform per-wave: full tile
    int k0  = half * 2;
    int row = base + l16;
    v2f a, b;
    a.x = x[(size_t)row * 5 + k0];
    a.y = x[(size_t)row * 5 + k0 + 1];
    int n = l16;
    b.x = W1[k0 * 16 + n];
    b.y = W1[(k0 + 1) * 16 + n];
    v8f c = {};
#if defined(__gfx1250__)
    c = __builtin_amdgcn_wmma_f32_16x16x4_f32(false, a, false, b, (short)0, c,
                                              false, false);
#endif
    float w4 = W1[4 * 16 + n];                 // K=4 correction column
    #pragma unroll
    for (int j = 0; j < 8; ++j) {
      int m = base + half * 8 + j;
      h1[(size_t)m * 16 + n] = c[j] + x[(size_t)m * 5 + 4] * w4;
    }
  } else {                                     // scalar tail (unused for N%16==0)
    if (half == 0) {
      int row = base + l16;
      if (row < N) {
        for (int n2 = 0; n2 < 16; ++n2) {
          float acc = 0.0f;
          for (int k = 0; k < 5; ++k) acc += x[(size_t)row * 5 + k] * W1[k * 16 + n2];
          h1[(size_t)row * 16 + n2] = acc;
        }
      }
    }
  }
}

// ---------------------------------------------------------- h2 = h1out @ W2 (16->4)
// Four chained f32 WMMAs (K = 16), B padded N 4->16 with zeros.
__global__ void k_hw2_wmma(const float* __restrict__ h, const float* __restrict__ W2,
                           float* __restrict__ h2, int N) {
  int lane = threadIdx.x & 31;
  int wave = (blockIdx.x * blockDim.x + threadIdx.x) >> 5;
  int base = wave * 16;
  if (base >= N) return;
  int half = lane >> 4;
  int l16  = lane & 15;
  if (base + 16 <= N) {
    int n = l16;
    int row = base + l16;
    v8f c = {};
    #pragma unroll
    for (int kk = 0; kk < 16; kk += 4) {
      int k0 = kk + half * 2;
      v2f a, b;
      a.x = h[(size_t)row * 16 + k0];
      a.y = h[(size_t)row * 16 + k0 + 1];
      b.x = (n < 4) ? W2[k0 * 4 + n] : 0.0f;
      b.y = (n < 4) ? W2[(k0 + 1) * 4 + n] : 0.0f;
#if defined(__gfx1250__)
      c = __builtin_amdgcn_wmma_f32_16x16x4_f32(false, a, false, b, (short)0, c,
                                                false, false);
#endif
    }
    if (n < 4) {
      #pragma unroll
      for (int j = 0; j < 8; ++j) {
        int m = base + half * 8 + j;
        h2[(size_t)m * 4 + n] = c[j];
      }
    }
  } else {
    if (half == 0) {
      int row = base + l16;
      if (row < N) {
        for (int n2 = 0; n2 < 4; ++n2) {
          float acc = 0.0f;
          for (int k = 0; k < 16; ++k) acc += h[(size_t)row * 16 + k] * W2[k * 4 + n2];
          h2[(size_t)row * 4 + n2] = acc;
        }
      }
    }
  }
}

// ------------------------------------------------- edge scatter: agg[dst] += h[src]*norm
// F consecutive lanes share one edge -> coalesced gathers & atomics.
template <int F, int LOGF>
__global__ void k_msg(const int* __restrict__ src, const int* __restrict__ dst,
                      const float* __restrict__ ea, const float* __restrict__ dis,
                      const float* __restrict__ h, float* __restrict__ agg, int E) {
  long gid = (long)blockIdx.x * blockDim.x + threadIdx.x;
  long e = gid >> LOGF;
  int  f = (int)(gid & (F - 1));
  if (e < E) {
    int s = src[e], t = dst[e];
    float nrm = dis[s] * ea[(size_t)e * 4] * dis[t];
    atomicAdd(&agg[(size_t)t * F + f], h[(size_t)s * F + f] * nrm);
  }
}

// out1 = relu(agg1 + h1*dis^2 (self loop) + b1), stored over agg1
__global__ void k_epi1(const float* __restrict__ h1, const float* __restrict__ dis,
                       const float* __restrict__ b1, float* __restrict__ agg1, int N) {
  long gid = (long)blockIdx.x * blockDim.x + threadIdx.x;
  int i = (int)(gid >> 4), f = (int)(gid & 15);
  if (i < N) {
    float ds = dis[i];
    float v = agg1[(size_t)i * 16 + f] + h1[(size_t)i * 16 + f] * ds * ds + b1[f];
    agg1[(size_t)i * 16 + f] = fmaxf(v, 0.0f);
  }
}

// out2 = agg2 + h2*dis^2 + b2 (no relu); block-reduce and atomically accumulate per-feature sums
__global__ void k_epi2(const float* __restrict__ h2, const float* __restrict__ dis,
                       const float* __restrict__ b2, const float* __restrict__ agg2,
                       float* __restrict__ acc, int N) {
  __shared__ float red[WG];
  long gid = (long)blockIdx.x * WG + threadIdx.x;
  int i = (int)(gid >> 2), f = (int)(gid & 3);
  float v = 0.0f;
  if (i < N) {
    float ds = dis[i];
    v = agg2[(size_t)i * 4 + f] + h2[(size_t)i * 4 + f] * ds * ds + b2[f];
  }
  red[threadIdx.x] = v;
  __syncthreads();
  for (int s = WG / 2; s >= 4; s >>= 1) {   // stride multiple of 4 keeps features separate
    if ((int)threadIdx.x < s) red[threadIdx.x] += red[threadIdx.x + s];
    __syncthreads();
  }
  if (threadIdx.x < 4) atomicAdd(&acc[threadIdx.x], red[threadIdx.x]);
}

// mean -> 4->128 relu -> 128->64 relu -> 64->50 -> softmax
__global__ void k_mlp(const float* __restrict__ acc,
                      const float* __restrict__ pW1, const float* __restrict__ pb1,
                      const float* __restrict__ pW2, const float* __restrict__ pb2,
                      const float* __restrict__ pW3, const float* __restrict__ pb3,
                      float* __restrict__ out, int N) {
  __shared__ float st[4];
  __shared__ float z1[128];
  __shared__ float z2[64];
  __shared__ float lg[50];
  int t = threadIdx.x;
  if (t < 4) st[t] = acc[t] / (float)N;
  __syncthreads();
  {
    float z = pb1[t];
    #pragma unroll
    for (int f = 0; f < 4; ++f) z += st[f] * pW1[f * 128 + t];
    z1[t] = fmaxf(z, 0.0f);
  }
  __syncthreads();
  if (t < 64) {
    float z = pb2[t];
    for (int j = 0; j < 128; ++j) z += z1[j] * pW2[j * 64 + t];
    z2[t] = fmaxf(z, 0.0f);
  }
  __syncthreads();
  if (t < 50) {
    float z = pb3[t];
    for (int k = 0; k < 64; ++k) z += z2[k] * pW3[k * 50 + t];
    lg[t] = z;
  }
  __syncthreads();
  if (t == 0) {
    float mx = -3.402823466e38f;
    for (int l = 0; l < 50; ++l) mx = fmaxf(mx, lg[l]);
    float s = 0.0f;
    for (int l = 0; l < 50; ++l) { float e = expf(lg[l] - mx); lg[l] = e; s += e; }
    float inv = 1.0f / s;
    for (int l = 0; l < 50; ++l) out[l] = lg[l] * inv;
  }
}

// ---------------------------------------------------------------- launcher
extern "C" void kernel_launch(void* const* d_in, const int* in_sizes, int n_in,
                              void* d_out, int out_size, void* d_ws, size_t ws_size,
                              hipStream_t stream) {
  const float* x   = (const float*)d_in[0];
  const int*   ei  = (const int*)  d_in[1];
  const float* ea  = (const float*)d_in[2];
  const float* W1  = (const float*)d_in[3];
  const float* b1  = (const float*)d_in[4];
  const float* W2  = (const float*)d_in[5];
  const float* b2  = (const float*)d_in[6];
  const float* pW1 = (const float*)d_in[7];
  const float* pb1 = (const float*)d_in[8];
  const float* pW2 = (const float*)d_in[9];
  const float* pb2 = (const float*)d_in[10];
  const float* pW3 = (const float*)d_in[11];
  const float* pb3 = (const float*)d_in[12];

  int N = in_sizes[0] / 5;
  int E = in_sizes[1] / 2;
  const int* srcI = ei;
  const int* dstI = ei + E;

  float* ws = (float*)d_ws;
  size_t off = 0;
  float* dis  = ws + off; off += (size_t)N;        // deg -> rsqrt(deg), in place
  float* h1   = ws + off; off += (size_t)N * 16;   // x @ W1
  float* agg1 = ws + off; off += (size_t)N * 16;   // layer-1 aggregate -> relu out
  float* h2   = ws + off; off += (size_t)N * 4;    // out1 @ W2
  float* agg2 = ws + off; off += (size_t)N * 4;    // layer-2 aggregate
  float* acc  = ws + off; off += 4;                // feature sums for mean

  auto cdiv = [](long a, long b) { return (int)((a + b - 1) / b); };
  long waves = (N + 15) / 16;

  k_fill<<<cdiv(N, WG), WG, 0, stream>>>(dis, N, 1.0f);            // self-loop deg
  k_fill<<<cdiv((long)N * 16, WG), WG, 0, stream>>>(agg1, N * 16, 0.0f);
  k_fill<<<cdiv((long)N * 4, WG), WG, 0, stream>>>(agg2, N * 4, 0.0f);
  k_fill<<<1, WG, 0, stream>>>(acc, 4, 0.0f);

  k_degree<<<cdiv(E, WG), WG, 0, stream>>>(dstI, ea, dis, E);
  k_rsqrt<<<cdiv(N, WG), WG, 0, stream>>>(dis, N);

  k_xw1_wmma<<<cdiv(waves * 32, WG), WG, 0, stream>>>(x, W1, h1, N);
  k_msg<16, 4><<<cdiv((long)E * 16, WG), WG, 0, stream>>>(srcI, dstI, ea, dis, h1, agg1, E);
  k_epi1<<<cdiv((long)N * 16, WG), WG, 0, stream>>>(h1, dis, b1, agg1, N);

  k_hw2_wmma<<<cdiv(waves * 32, WG), WG, 0, stream>>>(agg1, W2, h2, N);
  k_msg<4, 2><<<cdiv((long)E * 4, WG), WG, 0, stream>>>(srcI, dstI, ea, dis, h2, agg2, E);
  k_epi2<<<cdiv((long)N * 4, WG), WG, 0, stream>>>(h2, dis, b2, agg2, acc, N);

  k_mlp<<<1, 128, 0, stream>>>(acc, pW1, pb1, pW2, pb2, pW3, pb3, (float*)d_out, N);
}